// Ernie4_5_VLMoeMoeBlock_46746424050025
// MI455X (gfx1250) — compile-verified
//
#include <hip/hip_runtime.h>

typedef float v2f __attribute__((ext_vector_type(2)));
typedef float v4f __attribute__((ext_vector_type(4)));
typedef float v8f __attribute__((ext_vector_type(8)));

#define HIDDEN 1024
#define NEXP   16
#define T_TOK  1024

__device__ __forceinline__ v8f vzero8() {
  v8f z = {0.f, 0.f, 0.f, 0.f, 0.f, 0.f, 0.f, 0.f};
  return z;
}

// D = A(16x4) * B(4x16) + C, fp32 WMMA
__device__ __forceinline__ v8f wmma_k4(v2f a, v2f b, v8f c) {
  return __builtin_amdgcn_wmma_f32_16x16x4_f32(false, a, false, b, (short)0, c,
                                               false, false);
}

// Row-major LDS tile with 16-byte-block XOR swizzle:
//   element (r, c) lives at r*ld + (((c>>2) ^ r) << 2 | (c & 3))
// - async b128 copies (4 floats, block-aligned) stay contiguous
// - A-fragment loads (lanes 0..15 -> rows 0..15, fixed k) hit 16 distinct
//   4-bank blocks -> conflict-free; merged (k,k+1) pairs form ds_load_b64.
__device__ __forceinline__ int lds_idx(int r, int c, int ld) {
  return r * ld + (((((c >> 2) ^ r) & 0xFFFF) << 2) | (c & 3));
}

// global -> LDS async DMA, 16 bytes per lane (ASYNCcnt-tracked)
__device__ __forceinline__ void async_gather_b128(unsigned lds_byte_off,
                                                  const float* gptr) {
  asm volatile("global_load_async_to_lds_b128 %0, %1, off"
               :
               : "v"(lds_byte_off), "v"(gptr)
               : "memory");
}
__device__ __forceinline__ void wait_async0() {
  asm volatile("s_wait_asynccnt 0x0" ::: "memory");
}

// fast reciprocal: raw v_rcp_f32 (~1 ulp), avoids IEEE div expansion
__device__ __forceinline__ float frcp(float x) {
  return __builtin_amdgcn_rcpf(x);
}

__device__ __forceinline__ float silu_f(float g) {
  return g * frcp(1.f + __expf(-g));
}

__global__ void k_zero(int* p, int n) {
  int i = blockIdx.x * blockDim.x + threadIdx.x;
  if (i < n) p[i] = 0;
}

// ---------------------------------------------------------------------------
// Router: one wave per token. Computes 16 logits (chosen modality), softmax,
// top-2 on probs+bias, normalized weights. Appends (token|slot<<16, weight)
// to the per-expert dispatch lists.
// ---------------------------------------------------------------------------
__global__ __launch_bounds__(256) void k_router(
    const float* __restrict__ x, const int* __restrict__ tt,
    const float* __restrict__ rw_t, const float* __restrict__ bias_t,
    const float* __restrict__ rw_v, const float* __restrict__ bias_v,
    float* __restrict__ out_logits,
    int* __restrict__ cnt_t, int* __restrict__ tok_t, float* __restrict__ w_t,
    int* __restrict__ cnt_v, int* __restrict__ tok_v, float* __restrict__ w_v) {
  const int wave = threadIdx.x >> 5;
  const int lane = threadIdx.x & 31;
  const int t = blockIdx.x * 8 + wave;
  if (t >= T_TOK) return;
  const bool isv = (tt[t] != 0);
  const float* rw = isv ? rw_v : rw_t;
  const float* bias = isv ? bias_v : bias_t;

  float acc[NEXP];
#pragma unroll
  for (int e = 0; e < NEXP; ++e) acc[e] = 0.f;
  const float* xr = x + (size_t)t * HIDDEN;
  for (int h = lane; h < HIDDEN; h += 32) {
    float xv = xr[h];
#pragma unroll
    for (int e = 0; e < NEXP; ++e) acc[e] += xv * rw[e * HIDDEN + h];
  }
#pragma unroll
  for (int off = 16; off > 0; off >>= 1) {
#pragma unroll
    for (int e = 0; e < NEXP; ++e) acc[e] += __shfl_xor(acc[e], off, 32);
  }
  // softmax (all lanes redundantly)
  float mx = acc[0];
#pragma unroll
  for (int e = 1; e < NEXP; ++e) mx = fmaxf(mx, acc[e]);
  float p[NEXP], s = 0.f;
#pragma unroll
  for (int e = 0; e < NEXP; ++e) {
    p[e] = __expf(acc[e] - mx);
    s += p[e];
  }
  float inv = frcp(s);
#pragma unroll
  for (int e = 0; e < NEXP; ++e) p[e] *= inv;
  // top-2 of p+bias (earlier index wins ties, like lax.top_k)
  float s1 = -1e30f, s2 = -1e30f, w1 = 0.f, w2 = 0.f;
  int e1 = 0, e2 = 0;
#pragma unroll
  for (int e = 0; e < NEXP; ++e) {
    float v = p[e] + bias[e];
    if (v > s1) {
      s2 = s1; w2 = w1; e2 = e1;
      s1 = v;  w1 = p[e]; e1 = e;
    } else if (v > s2) {
      s2 = v; w2 = p[e]; e2 = e;
    }
  }
  float dnm = fmaxf(w1 + w2, 1e-12f);
  float rdnm = frcp(dnm);
  w1 *= rdnm;
  w2 *= rdnm;

  if (lane == 0) {
#pragma unroll
    for (int e = 0; e < NEXP; ++e) out_logits[t * NEXP + e] = acc[e];
    int* cnt = isv ? cnt_v : cnt_t;
    int* tok = isv ? tok_v : tok_t;
    float* wl = isv ? w_v : w_t;
    int p1 = atomicAdd(&cnt[e1], 1);
    tok[e1 * T_TOK + p1] = t;             // slot 0
    wl[e1 * T_TOK + p1] = w1;
    int p2 = atomicAdd(&cnt[e2], 1);
    tok[e2 * T_TOK + p2] = t | (1 << 16); // slot 1
    wl[e2 * T_TOK + p2] = w2;
  }
}

// ---------------------------------------------------------------------------
// Sparse expert MLP: block = (expert, 16-token tile). 8 waves.
// Stage 1: gu = X(16xH) @ gate_up[e](H x 2I); each wave owns matching g/u
//          column blocks -> register-local SiLU gating.
// Stage 2: y = h(16xI) @ down[e](I x H); write w*y to ys[slot][tok][:].
// ---------------------------------------------------------------------------
template <int I>
__global__ __launch_bounds__(256, 1) void k_expert(
    const float* __restrict__ x,
    const float* __restrict__ gate_up,  // [E][H][2I]
    const float* __restrict__ down,     // [E][I][H]
    const int* __restrict__ cnt, const int* __restrict__ tok,
    const float* __restrict__ wgt,
    float* __restrict__ ys) {           // [2][T][H]
  __shared__ __align__(16) float smem[16 * HIDDEN];  // 64 KB swizzled tile
  const int e = blockIdx.x;
  const int mt = blockIdx.y;
  const int c = cnt[e];
  if (mt * 16 >= c) return;
  const int tid = threadIdx.x;
  const int wave = tid >> 5;
  const int lane = tid & 31;
  const int m = lane & 15;
  const int n = lane & 15;
  const int kh = (lane >> 4) << 1;  // 0 or 2
  const int* tokp = tok + e * T_TOK + mt * 16;
  const float* wp = wgt + e * T_TOK + mt * 16;
  const int nrows = min(16, c - mt * 16);

  // gather up to 16 token rows of x into LDS via async b128 DMA
  const unsigned sbase = (unsigned)(uintptr_t)(&smem[0]);
  for (int i = tid; i < 16 * (HIDDEN / 4); i += 256) {
    int r = i >> 8;             // row 0..15
    int col = (i & 255) << 2;   // block-aligned column
    int idx = lds_idx(r, col, HIDDEN);
    if (r < nrows) {
      int tk = tokp[r] & 0xFFFF;
      async_gather_b128(sbase + 4u * (unsigned)idx,
                        x + (size_t)tk * HIDDEN + col);
    } else {
      v4f z = {0.f, 0.f, 0.f, 0.f};
      *(v4f*)&smem[idx] = z;
    }
  }
  wait_async0();
  __syncthreads();

  constexpr int NT1 = I / 128;  // 16-col tiles per (g,u) half per wave
  const float* gu = gate_up + (size_t)e * HIDDEN * (2 * I);
  v8f accg[NT1], accu[NT1];
#pragma unroll
  for (int j = 0; j < NT1; ++j) {
    accg[j] = vzero8();
    accu[j] = vzero8();
  }
  const int gbase = wave * (I / 8);
  for (int kb = 0; kb < HIDDEN; kb += 4) {
    const int kk = kb + kh;
    v2f a;
    a.x = smem[lds_idx(m, kk, HIDDEN)];
    a.y = smem[lds_idx(m, kk + 1, HIDDEN)];
    const float* b0 = gu + (size_t)kk * (2 * I);
    const float* b1 = b0 + (2 * I);
#pragma unroll
    for (int j = 0; j < NT1; ++j) {
      int cg = gbase + j * 16 + n;
      v2f bg;
      bg.x = b0[cg];
      bg.y = b1[cg];
      accg[j] = wmma_k4(a, bg, accg[j]);
      int cu = cg + I;
      v2f bu;
      bu.x = b0[cu];
      bu.y = b1[cu];
      accu[j] = wmma_k4(a, bu, accu[j]);
    }
  }
  __syncthreads();  // everyone done reading x tile
  // h = silu(g)*u -> LDS (row-major swizzled over I columns)
#pragma unroll
  for (int j = 0; j < NT1; ++j) {
#pragma unroll
    for (int r = 0; r < 8; ++r) {
      int mm = r + ((lane >> 4) << 3);
      int col = gbase + j * 16 + n;
      smem[lds_idx(mm, col, I)] = silu_f(accg[j][r]) * accu[j][r];
    }
  }
  __syncthreads();

  const float* dn = down + (size_t)e * I * HIDDEN;
  v8f acco[8];
#pragma unroll
  for (int j = 0; j < 8; ++j) acco[j] = vzero8();
  const int cb = wave * 128;
  for (int kb = 0; kb < I; kb += 4) {
    const int kk = kb + kh;
    v2f a;
    a.x = smem[lds_idx(m, kk, I)];
    a.y = smem[lds_idx(m, kk + 1, I)];
    const float* b0 = dn + (size_t)kk * HIDDEN;
    const float* b1 = b0 + HIDDEN;
#pragma unroll
    for (int j = 0; j < 8; ++j) {
      int cc = cb + j * 16 + n;
      v2f b;
      b.x = b0[cc];
      b.y = b1[cc];
      acco[j] = wmma_k4(a, b, acco[j]);
    }
  }
  // deterministic scatter: each (token, slot) row written exactly once
#pragma unroll
  for (int r = 0; r < 8; ++r) {
    int mm = r + ((lane >> 4) << 3);
    if (mm < nrows) {
      int ent = tokp[mm];
      int tk = ent & 0xFFFF;
      int slot = ent >> 16;
      float wv = wp[mm];
      float* dst = ys + (size_t)slot * T_TOK * HIDDEN + (size_t)tk * HIDDEN;
#pragma unroll
      for (int j = 0; j < 8; ++j) dst[cb + j * 16 + n] = wv * acco[j][r];
    }
  }
}

// ---------------------------------------------------------------------------
// Shared experts stage A: act = silu(x@Wg) * (x@Wu)   [T x 1024]
// grid = (64 m-tiles, 2 column halves); wave owns 64 cols of each half.
// ---------------------------------------------------------------------------
__global__ __launch_bounds__(256, 1) void k_shared_gu(
    const float* __restrict__ x, const float* __restrict__ wg,
    const float* __restrict__ wu, float* __restrict__ act) {
  __shared__ __align__(16) float smem[16 * HIDDEN];
  const int mt = blockIdx.x;
  const int nh = blockIdx.y;
  const int tid = threadIdx.x;
  const int wave = tid >> 5;
  const int lane = tid & 31;
  const int m = lane & 15;
  const int n = lane & 15;
  const int kh = (lane >> 4) << 1;

  const unsigned sbase = (unsigned)(uintptr_t)(&smem[0]);
  for (int i = tid; i < 16 * (HIDDEN / 4); i += 256) {
    int r = i >> 8;
    int col = (i & 255) << 2;
    async_gather_b128(sbase + 4u * (unsigned)lds_idx(r, col, HIDDEN),
                      x + (size_t)(mt * 16 + r) * HIDDEN + col);
  }
  wait_async0();
  __syncthreads();

  v8f ag[4], au[4];
#pragma unroll
  for (int j = 0; j < 4; ++j) {
    ag[j] = vzero8();
    au[j] = vzero8();
  }
  const int cb = nh * 512 + wave * 64;
  for (int kb = 0; kb < HIDDEN; kb += 4) {
    const int kk = kb + kh;
    v2f a;
    a.x = smem[lds_idx(m, kk, HIDDEN)];
    a.y = smem[lds_idx(m, kk + 1, HIDDEN)];
    const float* g0 = wg + (size_t)kk * HIDDEN;
    const float* g1 = g0 + HIDDEN;
    const float* u0 = wu + (size_t)kk * HIDDEN;
    const float* u1 = u0 + HIDDEN;
#pragma unroll
    for (int j = 0; j < 4; ++j) {
      int cc = cb + j * 16 + n;
      v2f bg;
      bg.x = g0[cc];
      bg.y = g1[cc];
      ag[j] = wmma_k4(a, bg, ag[j]);
      v2f bu;
      bu.x = u0[cc];
      bu.y = u1[cc];
      au[j] = wmma_k4(a, bu, au[j]);
    }
  }
#pragma unroll
  for (int j = 0; j < 4; ++j) {
#pragma unroll
    for (int r = 0; r < 8; ++r) {
      int mm = r + ((lane >> 4) << 3);
      act[(size_t)(mt * 16 + mm) * HIDDEN + cb + j * 16 + n] =
          silu_f(ag[j][r]) * au[j][r];
    }
  }
}

// ---------------------------------------------------------------------------
// Shared experts stage B + final combine:
//   out = act @ Wd + ys[slot0] + ys[slot1]
// ---------------------------------------------------------------------------
__global__ __launch_bounds__(256, 1) void k_shared_down_combine(
    const float* __restrict__ act, const float* __restrict__ wd,
    const float* __restrict__ ys, float* __restrict__ out) {
  __shared__ __align__(16) float smem[16 * HIDDEN];
  const int mt = blockIdx.x;
  const int nh = blockIdx.y;
  const int tid = threadIdx.x;
  const int wave = tid >> 5;
  const int lane = tid & 31;
  const int m = lane & 15;
  const int n = lane & 15;
  const int kh = (lane >> 4) << 1;

  const unsigned sbase = (unsigned)(uintptr_t)(&smem[0]);
  for (int i = tid; i < 16 * (HIDDEN / 4); i += 256) {
    int r = i >> 8;
    int col = (i & 255) << 2;
    async_gather_b128(sbase + 4u * (unsigned)lds_idx(r, col, HIDDEN),
                      act + (size_t)(mt * 16 + r) * HIDDEN + col);
  }
  wait_async0();
  __syncthreads();

  v8f ac[4];
#pragma unroll
  for (int j = 0; j < 4; ++j) ac[j] = vzero8();
  const int cb = nh * 512 + wave * 64;
  for (int kb = 0; kb < HIDDEN; kb += 4) {
    const int kk = kb + kh;
    v2f a;
    a.x = smem[lds_idx(m, kk, HIDDEN)];
    a.y = smem[lds_idx(m, kk + 1, HIDDEN)];
    const float* b0 = wd + (size_t)kk * HIDDEN;
    const float* b1 = b0 + HIDDEN;
#pragma unroll
    for (int j = 0; j < 4; ++j) {
      int cc = cb + j * 16 + n;
      v2f b;
      b.x = b0[cc];
      b.y = b1[cc];
      ac[j] = wmma_k4(a, b, ac[j]);
    }
  }
#pragma unroll
  for (int j = 0; j < 4; ++j) {
#pragma unroll
    for (int r = 0; r < 8; ++r) {
      int mm = r + ((lane >> 4) << 3);
      size_t idx = (size_t)(mt * 16 + mm) * HIDDEN + cb + j * 16 + n;
      out[idx] = ac[j][r] + ys[idx] + ys[idx + (size_t)T_TOK * HIDDEN];
    }
  }
}

extern "C" void kernel_launch(void* const* d_in, const int* in_sizes, int n_in,
                              void* d_out, int out_size, void* d_ws,
                              size_t ws_size, hipStream_t stream) {
  (void)in_sizes; (void)n_in; (void)out_size; (void)ws_size;
  const float* x   = (const float*)d_in[0];
  const int*   tt  = (const int*)d_in[1];
  const float* trw = (const float*)d_in[2];
  const float* tb  = (const float*)d_in[3];
  const float* tgu = (const float*)d_in[4];
  const float* tdn = (const float*)d_in[5];
  const float* vrw = (const float*)d_in[6];
  const float* vb  = (const float*)d_in[7];
  const float* vgu = (const float*)d_in[8];
  const float* vdn = (const float*)d_in[9];
  const float* sg  = (const float*)d_in[10];
  const float* su  = (const float*)d_in[11];
  const float* sd  = (const float*)d_in[12];

  float* out = (float*)d_out;
  float* out_logits = out + (size_t)T_TOK * HIDDEN;

  // workspace layout (all 4-byte elements)
  int* cnt_t = (int*)d_ws;
  int* cnt_v = cnt_t + 16;
  int* tok_t = cnt_v + 16;
  int* tok_v = tok_t + NEXP * T_TOK;
  float* w_t = (float*)(tok_v + NEXP * T_TOK);
  float* w_v = w_t + NEXP * T_TOK;
  float* act = w_v + NEXP * T_TOK;                 // [T][H] = 4 MB
  float* ys  = act + (size_t)T_TOK * HIDDEN;       // [2][T][H] = 8 MB

  k_zero<<<1, 64, 0, stream>>>(cnt_t, 32);
  k_router<<<128, 256, 0, stream>>>(x, tt, trw, tb, vrw, vb, out_logits, cnt_t,
                                    tok_t, w_t, cnt_v, tok_v, w_v);
  k_expert<512><<<dim3(16, 64), 256, 0, stream>>>(x, tgu, tdn, cnt_t, tok_t,
                                                  w_t, ys);
  k_expert<256><<<dim3(16, 64), 256, 0, stream>>>(x, vgu, vdn, cnt_v, tok_v,
                                                  w_v, ys);
  k_shared_gu<<<dim3(64, 2), 256, 0, stream>>>(x, sg, su, act);
  k_shared_down_combine<<<dim3(64, 2), 256, 0, stream>>>(act, sd, ys, out);
}